// MelNet_9002251452606
// MI455X (gfx1250) — compile-verified
//
#include <hip/hip_runtime.h>
#include <stdint.h>

// GMM per-pixel sampling, CDNA5 (gfx1250).
// Bandwidth-bound: ~554 MB moved -> ~24 us floor @ 23.3 TB/s.
// Uses the CDNA5 async-DMA path: global_load_async_to_lds_b128 (ASYNCcnt)
// with double-buffered LDS tiles; compute gathers from LDS.

#define GMM_K     10
#define TILE      256                      // pixels per tile
#define NTHREADS  256                      // 8 wave32 per block
#define KT_BYTES  (TILE * GMM_K * 4)       // 10240 B per K-tensor tile
#define S1_BYTES  (TILE * 4)               // 1024 B per scalar-tensor tile
#define BUF_BYTES (3 * KT_BYTES + 2 * S1_BYTES)   // 32768 B
#define NCHUNKS   (BUF_BYTES / 16)         // 2048 b128 chunks
#define CHUNKS_PER_THREAD (NCHUNKS / NTHREADS)    // exactly 8 -> uniform ASYNCcnt

// ---- CDNA5 async global -> LDS copy (tracked with ASYNCcnt) ----------------
__device__ __forceinline__ void async_copy_b128(uint32_t lds_off, const void* gsrc) {
  uint64_t a = (uint64_t)gsrc;
  asm volatile("global_load_async_to_lds_b128 %0, %1, off"
               :: "v"(lds_off), "v"(a)
               : "memory");
}

// Issue one tile's worth of async copies: 8 b128 ops per thread (uniform).
// LDS layout per buffer: [pi 10240][mu 10240][ls 10240][u 1024][eps 1024]
__device__ __forceinline__ void issue_tile(const float* __restrict__ pi,
                                           const float* __restrict__ mu,
                                           const float* __restrict__ ls,
                                           const float* __restrict__ uc,
                                           const float* __restrict__ ep,
                                           long long ts, uint32_t ldsBase,
                                           unsigned tid) {
  const char* pi_b = (const char*)(pi + ts * GMM_K);
  const char* mu_b = (const char*)(mu + ts * GMM_K);
  const char* ls_b = (const char*)(ls + ts * GMM_K);
  const char* uc_b = (const char*)(uc + ts);
  const char* ep_b = (const char*)(ep + ts);
  const unsigned C0 = KT_BYTES / 16;            // 640
  const unsigned C1 = 2u * C0;                  // 1280
  const unsigned C2 = 3u * C0;                  // 1920
  const unsigned C3 = C2 + S1_BYTES / 16;       // 1984
#pragma unroll
  for (int j = 0; j < CHUNKS_PER_THREAD; ++j) {
    unsigned g = tid + NTHREADS * (unsigned)j;  // region bounds are multiples
    const char* src; unsigned c;                // of 32 -> wave-uniform branch
    if (g < C0)      { src = pi_b; c = g; }
    else if (g < C1) { src = mu_b; c = g - C0; }
    else if (g < C2) { src = ls_b; c = g - C1; }
    else if (g < C3) { src = uc_b; c = g - C2; }
    else             { src = ep_b; c = g - C3; }
    async_copy_b128(ldsBase + g * 16u, src + (size_t)c * 16u);
  }
}

__device__ __forceinline__ void compute_tile(float* __restrict__ out, long long ts,
                                             const unsigned char* sbuf, unsigned tid) {
  const float* sp  = (const float*)(sbuf + (size_t)tid * (GMM_K * 4));
  const float* smu = (const float*)(sbuf + KT_BYTES);
  const float* sls = (const float*)(sbuf + 2 * KT_BYTES);
  const float* su  = (const float*)(sbuf + 3 * KT_BYTES);
  const float* se  = (const float*)(sbuf + 3 * KT_BYTES + S1_BYTES);

  float l[GMM_K];
#pragma unroll
  for (int j = 0; j < GMM_K; ++j) l[j] = sp[j];
  float m = l[0];
#pragma unroll
  for (int j = 1; j < GMM_K; ++j) m = fmaxf(m, l[j]);
  float e[GMM_K]; float S = 0.f;
#pragma unroll
  for (int j = 0; j < GMM_K; ++j) { e[j] = __expf(l[j] - m); S += e[j]; }
  // cdf_j < u  <=>  prefix_j < u * S   (S > 0)
  float thr = su[tid] * S;
  float acc = 0.f; int k = 0;
#pragma unroll
  for (int j = 0; j < GMM_K; ++j) { acc += e[j]; k += (acc < thr) ? 1 : 0; }
  if (k > GMM_K - 1) k = GMM_K - 1;
  float muk = smu[tid * GMM_K + (unsigned)k];
  float lsk = sls[tid * GMM_K + (unsigned)k];
  float r = fmaf(__expf(lsk), se[tid], muk);
  __builtin_nontemporal_store(r, &out[ts + tid]);
}

__global__ void __launch_bounds__(NTHREADS)
gmm_sample_pipelined(const float* __restrict__ mu, const float* __restrict__ ls,
                     const float* __restrict__ pi, const float* __restrict__ uc,
                     const float* __restrict__ ep, float* __restrict__ out,
                     int num_tiles) {
  __shared__ alignas(16) unsigned char smem[2][BUF_BYTES];  // 64 KB: 5 WGs/WGP
  int t = (int)blockIdx.x;
  if (t >= num_tiles) return;                 // block-uniform
  const int stride = (int)gridDim.x;
  const unsigned tid = threadIdx.x;
  // Low 32 bits of a generic pointer to LDS == wave-relative LDS byte offset.
  const uint32_t base0 = (uint32_t)(uintptr_t)(const void*)&smem[0][0];
  const uint32_t base1 = (uint32_t)(uintptr_t)(const void*)&smem[1][0];

  issue_tile(pi, mu, ls, uc, ep, (long long)t * TILE, base0, tid);
  int cb = 0;
  for (;;) {
    int nt = t + stride;
    const bool hasNext = nt < num_tiles;      // block-uniform
    if (hasNext) {
      issue_tile(pi, mu, ls, uc, ep, (long long)nt * TILE, cb ? base0 : base1, tid);
      // 8 ops/wave outstanding for 'next'; in-order completion => waiting to
      // <=8 guarantees the current tile's 8 ops have landed in LDS.
      asm volatile("s_wait_asynccnt 0x8" ::: "memory");
    } else {
      asm volatile("s_wait_asynccnt 0x0" ::: "memory");
    }
    __syncthreads();                          // all waves' copies visible
    compute_tile(out, (long long)t * TILE, cb ? smem[1] : smem[0], tid);
    if (!hasNext) break;
    __syncthreads();                          // done reading before next overwrite
    cb ^= 1;
    t = nt;
  }
}

// Fallback for a non-multiple-of-TILE tail (N is divisible for the given shapes).
__global__ void gmm_sample_tail(const float* __restrict__ mu, const float* __restrict__ ls,
                                const float* __restrict__ pi, const float* __restrict__ uc,
                                const float* __restrict__ ep, float* __restrict__ out,
                                long long start, long long n) {
  long long i = start + (long long)blockIdx.x * blockDim.x + threadIdx.x;
  if (i >= n) return;
  const float* p = pi + i * GMM_K;
  float l[GMM_K];
#pragma unroll
  for (int j = 0; j < GMM_K; ++j) l[j] = p[j];
  float m = l[0];
#pragma unroll
  for (int j = 1; j < GMM_K; ++j) m = fmaxf(m, l[j]);
  float e[GMM_K]; float S = 0.f;
#pragma unroll
  for (int j = 0; j < GMM_K; ++j) { e[j] = __expf(l[j] - m); S += e[j]; }
  float thr = uc[i] * S;
  float acc = 0.f; int k = 0;
#pragma unroll
  for (int j = 0; j < GMM_K; ++j) { acc += e[j]; k += (acc < thr) ? 1 : 0; }
  if (k > GMM_K - 1) k = GMM_K - 1;
  out[i] = fmaf(__expf(ls[i * GMM_K + k]), ep[i], mu[i * GMM_K + k]);
}

extern "C" void kernel_launch(void* const* d_in, const int* in_sizes, int n_in,
                              void* d_out, int out_size, void* d_ws, size_t ws_size,
                              hipStream_t stream) {
  (void)n_in; (void)d_ws; (void)ws_size; (void)in_sizes;
  const float* mu = (const float*)d_in[0];
  const float* ls = (const float*)d_in[1];
  const float* pi = (const float*)d_in[2];
  const float* uc = (const float*)d_in[3];
  const float* ep = (const float*)d_in[4];
  float* out = (float*)d_out;

  const long long N = (long long)out_size;    // B*F*T
  const int tiles = (int)(N / TILE);
  if (tiles > 0) {
    int blocks = tiles < 2048 ? tiles : 2048; // persistent, 8 tiles/block pipelined
    gmm_sample_pipelined<<<blocks, NTHREADS, 0, stream>>>(mu, ls, pi, uc, ep, out, tiles);
  }
  const long long tail_start = (long long)tiles * TILE;
  if (N > tail_start) {
    int tb = (int)((N - tail_start + 255) / 256);
    gmm_sample_tail<<<tb, 256, 0, stream>>>(mu, ls, pi, uc, ep, out, tail_start, N);
  }
}